// Decoder_17884243821375
// MI455X (gfx1250) — compile-verified
//
#include <hip/hip_runtime.h>
#include <math.h>

typedef float v2f __attribute__((ext_vector_type(2)));
typedef float v8f __attribute__((ext_vector_type(8)));

// Problem constants (from reference)
#define BB   32
#define TT   512
#define DD   1024
#define HH   1024
#define AA   512
#define V3N  8003
#define TDN  100
#define BOSN 8000
#define NEGF (-3.402823466e38f)

// ---------------- WMMA helpers ----------------
// One wave computes NT adjacent 16x16 f32 tiles sharing the same A rows:
// A fragment loaded once per k-chunk, NT independent accumulator chains.
// Fragment layouts per ISA 7.12.2:
//  A (16x4 f32): lanes 0-15 rows M=0..15 hold K=k+0,k+1; lanes 16-31 hold K=k+2,k+3.
//  B (4x16 f32): lane's column = lane&15; lane half selects K rows (symmetric).
//  D (16x16 f32): VGPR r, lane l<16 -> (M=r, N=l); lane>=16 -> (M=r+8, N=l-16).

template <int NT>
static __device__ __forceinline__ void wmma_AB_multi(const float* Ap, int lda,
                                                     const float* Bp, int ldb,
                                                     int K, v8f acc[NT]) {
  const int lane = threadIdx.x & 31;
  const int hf = lane >> 4, l16 = lane & 15;
  const float* ar = Ap + (size_t)l16 * lda;
  for (int k = 0; k < K; k += 4) {
    v2f a;
    a.x = ar[k + 2 * hf];
    a.y = ar[k + 2 * hf + 1];
#pragma unroll
    for (int i = 0; i < NT; ++i) {
      v2f b;
      b.x = Bp[(size_t)(k + 2 * hf) * ldb + i * 16 + l16];
      b.y = Bp[(size_t)(k + 2 * hf + 1) * ldb + i * 16 + l16];
      acc[i] = __builtin_amdgcn_wmma_f32_16x16x4_f32(false, a, false, b,
                                                     (short)0, acc[i],
                                                     false, false);
    }
  }
}

// B given as row-major weight W[N x K]; computes X @ W^T for NT adjacent tiles.
template <int NT>
static __device__ __forceinline__ void wmma_ABt_multi(const float* Ap, int lda,
                                                      const float* Wp, int ldw,
                                                      int K, v8f acc[NT]) {
  const int lane = threadIdx.x & 31;
  const int hf = lane >> 4, l16 = lane & 15;
  const float* ar = Ap + (size_t)l16 * lda;
  const float* wr[NT];
#pragma unroll
  for (int i = 0; i < NT; ++i) wr[i] = Wp + (size_t)(i * 16 + l16) * ldw;
  for (int k = 0; k < K; k += 4) {
    v2f a;
    a.x = ar[k + 2 * hf];
    a.y = ar[k + 2 * hf + 1];
#pragma unroll
    for (int i = 0; i < NT; ++i) {
      v2f b;
      b.x = wr[i][k + 2 * hf];
      b.y = wr[i][k + 2 * hf + 1];
      acc[i] = __builtin_amdgcn_wmma_f32_16x16x4_f32(false, a, false, b,
                                                     (short)0, acc[i],
                                                     false, false);
    }
  }
}

static __device__ __forceinline__ void wmma_store(float* Cp, int ldc, v8f acc) {
  const int lane = threadIdx.x & 31;
  const int hf = lane >> 4, l16 = lane & 15;
#pragma unroll
  for (int r = 0; r < 8; ++r)
    Cp[(size_t)(r + 8 * hf) * ldc + l16] = acc[r];
}

// ---------------- Kernels ----------------

// enc_proj[B*T, A] = enc_pad[B*T, D] @ W_enc_att[D, A]   (once)
// grid (1024, 8): wave mt handles 4 adjacent N tiles.
__global__ __launch_bounds__(32) void k_encproj(const float* __restrict__ enc,
                                                const float* __restrict__ Wenc,
                                                float* __restrict__ encproj) {
  const int mt = blockIdx.x;           // 0..1023
  const int n0 = blockIdx.y * 64;      // 4 tiles of 16
  v8f acc[4] = {};
  wmma_AB_multi<4>(enc + (size_t)(mt * 16) * DD, DD, Wenc + n0, AA, DD, acc);
#pragma unroll
  for (int i = 0; i < 4; ++i)
    wmma_store(encproj + (size_t)(mt * 16) * AA + n0 + i * 16, AA, acc[i]);
}

// init z,c from dec_z0, dec_c0
__global__ __launch_bounds__(256) void k_init(const float* __restrict__ z0,
                                              const float* __restrict__ c0,
                                              float* __restrict__ z,
                                              float* __restrict__ c) {
  int i = blockIdx.x * 256 + threadIdx.x;
  if (i < BB * HH) { z[i] = z0[i]; c[i] = c0[i]; }
}

// q[32, A] = z[32, H] @ W_dec_att[H, A]      grid (2, 8)
__global__ __launch_bounds__(32) void k_q(const float* __restrict__ z,
                                          const float* __restrict__ Wdec,
                                          float* __restrict__ q) {
  const int mt = blockIdx.x;       // 0..1
  const int n0 = blockIdx.y * 64;  // 4 tiles
  v8f acc[4] = {};
  wmma_AB_multi<4>(z + (size_t)(mt * 16) * HH, HH, Wdec + n0, AA, HH, acc);
#pragma unroll
  for (int i = 0; i < 4; ++i)
    wmma_store(q + (size_t)(mt * 16) * AA + n0 + i * 16, AA, acc[i]);
}

// e[b,t] = sum_a tanh(enc_proj[b,t,a] + q[b,a]) * v_att[a]  (masked)
__global__ __launch_bounds__(256) void k_att_e(const float* __restrict__ encproj,
                                               const float* __restrict__ q,
                                               const float* __restrict__ vatt,
                                               const int* __restrict__ enc_len,
                                               float* __restrict__ e) {
  const int wid = (blockIdx.x * 256 + threadIdx.x) >> 5;  // 0..16383
  const int lane = threadIdx.x & 31;
  const int b = wid >> 9;
  const int t = wid & (TT - 1);
  const float* ep = encproj + ((size_t)b * TT + t) * AA;
  const float* qq = q + (size_t)b * AA;
  float acc = 0.f;
  for (int a = lane; a < AA; a += 32) acc += tanhf(ep[a] + qq[a]) * vatt[a];
#pragma unroll
  for (int m = 16; m; m >>= 1) acc += __shfl_xor(acc, m, 32);
  if (lane == 0) e[b * TT + t] = (t < enc_len[b]) ? acc : NEGF;
}

// masked softmax over T per batch row; writes ws scratch + ws_all output slice
__global__ __launch_bounds__(512) void k_softmax(const float* __restrict__ e,
                                                 float* __restrict__ ws,
                                                 float* __restrict__ out_wsall,
                                                 int step) {
  __shared__ float redA[16];
  __shared__ float redB[16];
  const int b = blockIdx.x, t = threadIdx.x;
  const int wid = t >> 5, lane = t & 31;
  float v = e[b * TT + t];
  float m = v;
#pragma unroll
  for (int s = 16; s; s >>= 1) m = fmaxf(m, __shfl_xor(m, s, 32));
  if (lane == 0) redA[wid] = m;
  __syncthreads();
  if (t < 16) {
    float x = redA[t];
#pragma unroll
    for (int s = 8; s; s >>= 1) x = fmaxf(x, __shfl_xor(x, s, 32));
    if (t == 0) redA[0] = x;
  }
  __syncthreads();
  const float M = redA[0];
  float ex = expf(v - M);
  float su = ex;
#pragma unroll
  for (int s = 16; s; s >>= 1) su += __shfl_xor(su, s, 32);
  if (lane == 0) redB[wid] = su;
  __syncthreads();
  if (t < 16) {
    float x = redB[t];
#pragma unroll
    for (int s = 8; s; s >>= 1) x += __shfl_xor(x, s, 32);
    if (t == 0) redB[0] = x;
  }
  __syncthreads();
  const float w = ex / redB[0];
  ws[b * TT + t] = w;
  out_wsall[((size_t)b * TDN + step) * TT + t] = w;
}

// ctx[b,d] = sum_t ws[b,t] * enc_pad[b,t,d]
__global__ __launch_bounds__(128) void k_ctx(const float* __restrict__ ws,
                                             const float* __restrict__ enc,
                                             float* __restrict__ ctx) {
  __shared__ float w[TT];
  const int b = blockIdx.x;
  const int d = blockIdx.y * 128 + threadIdx.x;
  for (int t = threadIdx.x; t < TT; t += 128) w[t] = ws[b * TT + t];
  __syncthreads();
  const float* ep = enc + (size_t)b * TT * DD + d;
  float acc = 0.f;
  for (int t = 0; t < TT; ++t) acc += w[t] * ep[(size_t)t * DD];
  ctx[b * DD + d] = acc;
}

// x[b,:] = concat(emb_table[prev_tok[b]], ctx[b,:])
__global__ __launch_bounds__(256) void k_x(const float* __restrict__ emb,
                                           const float* __restrict__ ctx,
                                           const int* __restrict__ preds,
                                           float* __restrict__ x, int step) {
  const int idx = blockIdx.x * 256 + threadIdx.x;  // 0..65535
  const int b = idx >> 11;
  const int j = idx & 2047;
  const int prev = (step == 0) ? BOSN : preds[b * TDN + step - 1];
  x[idx] = (j < HH) ? emb[(size_t)prev * HH + j] : ctx[b * DD + (j - HH)];
}

// gates[32,4096] = x @ W_ih^T + z @ W_hh^T + b_ih + b_hh    grid (2, 64)
__global__ __launch_bounds__(32) void k_gates(const float* __restrict__ x,
                                              const float* __restrict__ z,
                                              const float* __restrict__ Wih,
                                              const float* __restrict__ Whh,
                                              const float* __restrict__ bih,
                                              const float* __restrict__ bhh,
                                              float* __restrict__ gates) {
  const int mt = blockIdx.x;           // 0..1
  const int n0 = blockIdx.y * 64;      // 4 tiles of 16
  v8f acc[4] = {};
  wmma_ABt_multi<4>(x + (size_t)(mt * 16) * (HH + DD), HH + DD,
                    Wih + (size_t)n0 * (HH + DD), HH + DD, HH + DD, acc);
  wmma_ABt_multi<4>(z + (size_t)(mt * 16) * HH, HH,
                    Whh + (size_t)n0 * HH, HH, HH, acc);
  const int lane = threadIdx.x & 31;
  const int hf = lane >> 4, l16 = lane & 15;
#pragma unroll
  for (int i = 0; i < 4; ++i) {
    const int n = n0 + i * 16 + l16;
    const float bias = bih[n] + bhh[n];
#pragma unroll
    for (int r = 0; r < 8; ++r)
      gates[(size_t)(mt * 16 + r + 8 * hf) * (4 * HH) + n] = acc[i][r] + bias;
  }
}

__device__ __forceinline__ float sigm(float v) { return 1.f / (1.f + expf(-v)); }

// LSTM elementwise; updates z, c in place
__global__ __launch_bounds__(256) void k_lstm(const float* __restrict__ gates,
                                              float* __restrict__ z,
                                              float* __restrict__ c) {
  const int idx = blockIdx.x * 256 + threadIdx.x;  // 0..32767
  const int b = idx >> 10, h = idx & (HH - 1);
  const float* g = gates + (size_t)b * (4 * HH);
  const float gi = g[h], gf = g[HH + h], gg = g[2 * HH + h], go = g[3 * HH + h];
  const float cn = sigm(gf) * c[idx] + sigm(gi) * tanhf(gg);
  const float zn = sigm(go) * tanhf(cn);
  c[idx] = cn;
  z[idx] = zn;
}

// logits[b, step, :] = z @ W_out^T + b_out   grid (2, 126): 504 tiles cover 501
__global__ __launch_bounds__(32) void k_logits(const float* __restrict__ z,
                                               const float* __restrict__ Wout,
                                               const float* __restrict__ bout,
                                               float* __restrict__ logits,
                                               int step) {
  const int mt = blockIdx.x;        // 0..1
  const int n0 = blockIdx.y * 64;   // 4 tiles of 16
  const int lane = threadIdx.x & 31;
  const int hf = lane >> 4, l16 = lane & 15;

  v8f acc[4] = {};
  {
    // clamped B row pointers: OOB columns compute garbage but are never stored
    const float* ar = z + (size_t)(mt * 16) * HH + (size_t)l16 * HH;
    const float* wr[4];
#pragma unroll
    for (int i = 0; i < 4; ++i) {
      int n = n0 + i * 16 + l16;
      int nc = (n < V3N) ? n : (V3N - 1);
      wr[i] = Wout + (size_t)nc * HH;
    }
    for (int k = 0; k < HH; k += 4) {
      v2f a;
      a.x = ar[k + 2 * hf];
      a.y = ar[k + 2 * hf + 1];
#pragma unroll
      for (int i = 0; i < 4; ++i) {
        v2f b;
        b.x = wr[i][k + 2 * hf];
        b.y = wr[i][k + 2 * hf + 1];
        acc[i] = __builtin_amdgcn_wmma_f32_16x16x4_f32(false, a, false, b,
                                                       (short)0, acc[i],
                                                       false, false);
      }
    }
  }
#pragma unroll
  for (int i = 0; i < 4; ++i) {
    const int n = n0 + i * 16 + l16;
    if (n < V3N) {
      const float bias = bout[n];
#pragma unroll
      for (int r = 0; r < 8; ++r) {
        const int m = mt * 16 + r + 8 * hf;  // batch index
        logits[((size_t)m * TDN + step) * V3N + n] = acc[i][r] + bias;
      }
    }
  }
}

// argmax over V3 per batch (first occurrence on ties, as jnp.argmax)
__global__ __launch_bounds__(256) void k_argmax(const float* __restrict__ logits,
                                                int* __restrict__ preds, int step) {
  __shared__ float sv[256];
  __shared__ int si[256];
  const int b = blockIdx.x, tid = threadIdx.x;
  const float* row = logits + ((size_t)b * TDN + step) * V3N;
  float best = NEGF;
  int bi = 0;
  for (int v = tid; v < V3N; v += 256) {
    const float x = row[v];
    if (x > best) { best = x; bi = v; }
  }
  sv[tid] = best; si[tid] = bi;
  __syncthreads();
  for (int s = 128; s; s >>= 1) {
    if (tid < s) {
      if (sv[tid + s] > sv[tid] ||
          (sv[tid + s] == sv[tid] && si[tid + s] < si[tid])) {
        sv[tid] = sv[tid + s]; si[tid] = si[tid + s];
      }
    }
    __syncthreads();
  }
  if (tid == 0) preds[b * TDN + step] = si[0];
}

// lse[b,v] = logsumexp over t of logits[b,t,v]   (log_softmax over time axis)
__global__ __launch_bounds__(256) void k_lse(const float* __restrict__ logits,
                                             float* __restrict__ lse) {
  const int idx = blockIdx.x * 256 + threadIdx.x;
  if (idx >= BB * V3N) return;
  const int b = idx / V3N, v = idx % V3N;
  const float* p = logits + (size_t)b * TDN * V3N + v;
  float m = NEGF;
  for (int t = 0; t < TDN; ++t) m = fmaxf(m, p[(size_t)t * V3N]);
  float s = 0.f;
  for (int t = 0; t < TDN; ++t) s += expf(p[(size_t)t * V3N] - m);
  lse[idx] = m + logf(s);
}

// ys_log_probs[b,t] = logits[b,t,p] - lse[b,p];  also emit preds as float
__global__ __launch_bounds__(256) void k_final(const float* __restrict__ logits,
                                               const float* __restrict__ lse,
                                               const int* __restrict__ preds,
                                               float* __restrict__ out) {
  const int idx = blockIdx.x * 256 + threadIdx.x;
  if (idx >= BB * TDN) return;
  const int b = idx / TDN, t = idx % TDN;
  const int p = preds[idx];
  out[idx] = logits[((size_t)b * TDN + t) * V3N + p] - lse[(size_t)b * V3N + p];
  out[BB * TDN + idx] = (float)p;
}

// ---------------- Host side ----------------
extern "C" void kernel_launch(void* const* d_in, const int* in_sizes, int n_in,
                              void* d_out, int out_size, void* d_ws, size_t ws_size,
                              hipStream_t stream) {
  (void)in_sizes; (void)n_in; (void)out_size; (void)ws_size;
  const float* enc_pad  = (const float*)d_in[0];
  const int*   enc_len  = (const int*)  d_in[1];
  const float* dec_z0   = (const float*)d_in[2];
  const float* dec_c0   = (const float*)d_in[3];
  const float* emb      = (const float*)d_in[4];
  const float* Wenc     = (const float*)d_in[5];
  const float* Wdec     = (const float*)d_in[6];
  const float* vatt     = (const float*)d_in[7];
  const float* Wih      = (const float*)d_in[8];
  const float* bih      = (const float*)d_in[9];
  const float* Whh      = (const float*)d_in[10];
  const float* bhh      = (const float*)d_in[11];
  const float* Wout     = (const float*)d_in[12];
  const float* bout     = (const float*)d_in[13];
  // d_in[14] = max_dec_timesteps (known constant 100)

  float* out = (float*)d_out;
  float* wsf = (float*)d_ws;

  // scratch layout (float elements)
  float* encproj = wsf;                        // 32*512*512   = 8388608
  float* q       = encproj + 8388608;          // 32*512       = 16384
  float* e       = q + 16384;                  // 32*512
  float* attw    = e + 16384;                  // 32*512
  float* ctx     = attw + 16384;               // 32*1024      = 32768
  float* x       = ctx + 32768;                // 32*2048      = 65536
  float* z       = x + 65536;                  // 32*1024
  float* c       = z + 32768;                  // 32*1024
  float* gates   = c + 32768;                  // 32*4096      = 131072
  float* lse     = gates + 131072;             // 32*8003      = 256096
  float* logits  = lse + 256096;               // 32*100*8003  = 25609600
  int*   preds   = (int*)(logits + 25609600);  // 32*100

  float* out_wsall = out + BB * TDN * 2;       // ws_all after ys_log_probs, preds

  // one-time: encoder projection + state init
  k_encproj<<<dim3(1024, 8), 32, 0, stream>>>(enc_pad, Wenc, encproj);
  k_init<<<128, 256, 0, stream>>>(dec_z0, dec_c0, z, c);

  for (int step = 0; step < TDN; ++step) {
    k_q      <<<dim3(2, 8),   32,  0, stream>>>(z, Wdec, q);
    k_att_e  <<<2048,         256, 0, stream>>>(encproj, q, vatt, enc_len, e);
    k_softmax<<<BB,           512, 0, stream>>>(e, attw, out_wsall, step);
    k_ctx    <<<dim3(BB, 8),  128, 0, stream>>>(attw, enc_pad, ctx);
    k_x      <<<256,          256, 0, stream>>>(emb, ctx, preds, x, step);
    k_gates  <<<dim3(2, 64),  32,  0, stream>>>(x, z, Wih, Whh, bih, bhh, gates);
    k_lstm   <<<128,          256, 0, stream>>>(gates, z, c);
    k_logits <<<dim3(2, 126), 32,  0, stream>>>(z, Wout, bout, logits, step);
    k_argmax <<<BB,           256, 0, stream>>>(logits, preds, step);
  }

  k_lse  <<<(BB * V3N + 255) / 256, 256, 0, stream>>>(logits, lse);
  k_final<<<(BB * TDN + 255) / 256, 256, 0, stream>>>(logits, lse, preds, out);
}